// QuanvolutionFilterQML_65481071402252
// MI455X (gfx1250) — compile-verified
//
#include <hip/hip_runtime.h>

// QuanvolutionFilterQML, closed form.
//
// Reference reduces exactly to (see derivation in commit message / analysis):
//   t0 = x[b, 2i, 2j  ] / 255 * (pi/2)
//   t1 = x[b, 2i, 2j+1] / 255 * (pi/2)
//   out[b, (i*128 + j)*2 + 0] = cos(t0) * cos(t1)
//   out[b, (i*128 + j)*2 + 1] = cos(t0)
// (the 8 "layers" are amplitude permutations + pure phases; |.|^2 kills the
//  phases and the net permutation yields z0 = cos t0 cos t1, z1 = cos t0).
//
// Memory-bound streaming map:
//   - only even rows of x are read: 64 MiB
//   - output: 64 MiB
//   -> ~5.5 us at 23.3 TB/s HBM; 128 MiB working set also fits the 192 MB L2,
//      so graph replays run mostly from L2. Keep default (RT) cache policy.
//
// Compacted 1:1 layout: even-row r of the input (256 floats at r*512) maps to
// output floats [r*256, r*256+256). Each thread handles one float4 (2 patches):
// coalesced b128 load, 4x v_cos_f32 (input pre-scaled to revolutions, range
// [0,0.25] so HW cos is exact enough, no range reduction), coalesced b128 store.

__global__ __launch_bounds__(256) void quanv_closed_form(
    const float* __restrict__ x,   // (512,1,256,256) f32
    float*       __restrict__ out, // (512, 32768)    f32
    unsigned n4)                   // total float4 outputs
{
    unsigned f = blockIdx.x * blockDim.x + threadIdx.x;   // float4 index
    if (f >= n4) return;

    unsigned row = f >> 6;   // even-row index across all 512 images (0..65535)
    unsigned col = f & 63u;  // float4 within the 256-float row

    // input: even row r lives at float offset r*512 (skip odd rows)
    const float4 p =
        reinterpret_cast<const float4*>(x + (size_t)row * 512u)[col];

    // theta = p/255 * pi/2 ; v_cos_f32 takes revolutions: theta/(2pi) = p/1020
    const float s = 1.0f / 1020.0f;
    float c0 = __builtin_amdgcn_cosf(p.x * s);
    float c1 = __builtin_amdgcn_cosf(p.y * s);
    float c2 = __builtin_amdgcn_cosf(p.z * s);
    float c3 = __builtin_amdgcn_cosf(p.w * s);

    float4 o;
    o.x = c0 * c1;   // z0 of patch (even col pair)
    o.y = c0;        // z1
    o.z = c2 * c3;   // z0 of next patch
    o.w = c2;        // z1
    reinterpret_cast<float4*>(out)[f] = o;
}

extern "C" void kernel_launch(void* const* d_in, const int* in_sizes, int n_in,
                              void* d_out, int out_size, void* d_ws, size_t ws_size,
                              hipStream_t stream) {
    (void)in_sizes; (void)n_in; (void)d_ws; (void)ws_size;
    const float* x = (const float*)d_in[0];
    float* out = (float*)d_out;

    unsigned n4 = (unsigned)(out_size / 4);          // 4,194,304
    unsigned blocks = (n4 + 255u) / 256u;            // 16,384
    quanv_closed_form<<<blocks, 256, 0, stream>>>(x, out, n4);
}